// CoordinationAnalysis_24730421690753
// MI455X (gfx1250) — compile-verified
//
#include <hip/hip_runtime.h>
#include <math.h>

// Problem constants (from reference): N=8192, D=256, S=3 shells
#define N_PTS 8192
#define DIM   256
#define NSH   3
#define LDSTRIDE 260   // 256 + 4 pad floats -> rows start 4 banks apart, conflict-free

typedef __attribute__((ext_vector_type(2))) float v2f;
typedef __attribute__((ext_vector_type(8))) float v8f;

// ---------------------------------------------------------------------------
// Kernel 1: row squared norms. One wave32 per row, 2 float4 per lane.
// ---------------------------------------------------------------------------
__global__ __launch_bounds__(256) void sq_kernel(const float* __restrict__ x,
                                                 float* __restrict__ sq) {
  int gwarp = (blockIdx.x * blockDim.x + threadIdx.x) >> 5;   // row index
  int lane  = threadIdx.x & 31;
  const float4* xr = (const float4*)(x + (size_t)gwarp * DIM);
  float s = 0.f;
#pragma unroll
  for (int q = 0; q < 2; ++q) {
    float4 v = xr[lane + 32 * q];
    s += v.x * v.x + v.y * v.y + v.z * v.z + v.w * v.w;
  }
#pragma unroll
  for (int m = 16; m >= 1; m >>= 1) s += __shfl_xor(s, m, 32);
  if (lane == 0) sq[gwarp] = s;
}

// ---------------------------------------------------------------------------
// Kernel 2: Gram-matrix sweep via V_WMMA_F32_16X16X4_F32 + shell counting.
// Grid: N/128 workgroups of 256 threads (8 waves). Wave w owns rows
// [blk*128 + w*16, +16). A strip held in registers in WMMA A layout.
// Column slabs of 32 staged through double-buffered LDS.
// ---------------------------------------------------------------------------
__global__ __launch_bounds__(256) void gram_shell_kernel(
    const float* __restrict__ x, const float* __restrict__ sq,
    const float* __restrict__ bnd, float* __restrict__ occ_out) {
  __shared__ float Bt[2][32 * LDSTRIDE];

  const int tid  = threadIdx.x;
  const int lane = tid & 31;
  const int wave = tid >> 5;
  const int half = lane >> 4;   // 0: K+0..1 , 1: K+2..3 (A/B layout)
  const int r    = lane & 15;   // M (for A) / N (for B) within tile
  const int row0 = blockIdx.x * 128 + wave * 16;

  // softplus(shell_boundaries) -> 4 bounds (3 shells)
  const float l0 = log1pf(expf(bnd[0]));
  const float l1 = log1pf(expf(bnd[1]));
  const float l2 = log1pf(expf(bnd[2]));
  const float u2 = log1pf(expf(bnd[3]));

  // A strip in registers, native WMMA A layout: a[kc] = x[row0+r][4kc+2h .. +1]
  v2f a[DIM / 4];
  {
    const float* arow = x + (size_t)(row0 + r) * DIM + 2 * half;
#pragma unroll
    for (int kc = 0; kc < DIM / 4; ++kc)
      a[kc] = *(const v2f*)(arow + 4 * kc);
  }

  // sq for this wave's 8 C-rows per half (C layout: M = v + 8*half)
  float sqA[8];
#pragma unroll
  for (int v = 0; v < 8; ++v) sqA[v] = sq[row0 + v + 8 * half];

  float cnt[8][NSH];
#pragma unroll
  for (int v = 0; v < 8; ++v)
#pragma unroll
    for (int s = 0; s < NSH; ++s) cnt[v][s] = 0.f;

  const int fill_row = tid >> 3;         // 0..31
  const int fill_col = (tid & 7) * 32;   // 8 float4 each

  for (int jb = 0; jb < N_PTS / 32; ++jb) {
    const int jbase = jb * 32;
    float* buf = &Bt[jb & 1][0];

    // ---- cooperative fill of 32x256 slab (coalesced b128 loads) ----
    {
      const float4* src = (const float4*)(x + (size_t)(jbase + fill_row) * DIM + fill_col);
      float4* dst = (float4*)(buf + fill_row * LDSTRIDE + fill_col);
#pragma unroll
      for (int q = 0; q < 8; ++q) dst[q] = src[q];
    }
    __syncthreads();   // single barrier/iter is safe with double buffering

    // ---- two 16x16 Gram tiles, K=256 in chunks of 4 ----
    v8f c0 = {}, c1 = {};
    const float* b0p = buf + r * LDSTRIDE + 2 * half;
    const float* b1p = buf + (16 + r) * LDSTRIDE + 2 * half;
#pragma unroll
    for (int kc = 0; kc < DIM / 4; ++kc) {
      v2f b0 = *(const v2f*)(b0p + 4 * kc);
      v2f b1 = *(const v2f*)(b1p + 4 * kc);
      c0 = __builtin_amdgcn_wmma_f32_16x16x4_f32(false, a[kc], false, b0,
                                                 (short)0, c0, false, false);
      c1 = __builtin_amdgcn_wmma_f32_16x16x4_f32(false, a[kc], false, b1,
                                                 (short)0, c1, false, false);
    }

    // ---- shell counting epilogue ----
    const float sqB0 = sq[jbase + r];
    const float sqB1 = sq[jbase + 16 + r];
#pragma unroll
    for (int v = 0; v < 8; ++v) {
      float d2 = sqA[v] + sqB0 - 2.f * c0[v];
      float t  = sqrtf(fmaxf(d2, 0.f));
      cnt[v][0] += (t >= l0 && t < l1) ? 1.f : 0.f;
      cnt[v][1] += (t >= l1 && t < l2) ? 1.f : 0.f;
      cnt[v][2] += (t >= l2 && t < u2) ? 1.f : 0.f;

      d2 = sqA[v] + sqB1 - 2.f * c1[v];
      t  = sqrtf(fmaxf(d2, 0.f));
      cnt[v][0] += (t >= l0 && t < l1) ? 1.f : 0.f;
      cnt[v][1] += (t >= l1 && t < l2) ? 1.f : 0.f;
      cnt[v][2] += (t >= l2 && t < u2) ? 1.f : 0.f;
    }
    // no trailing barrier needed: next iter writes the other buffer
  }

  // reduce across the 16 lanes sharing each M-row (masks stay within half)
#pragma unroll
  for (int v = 0; v < 8; ++v)
#pragma unroll
    for (int s = 0; s < NSH; ++s) {
      float t = cnt[v][s];
      t += __shfl_xor(t, 1, 32);
      t += __shfl_xor(t, 2, 32);
      t += __shfl_xor(t, 4, 32);
      t += __shfl_xor(t, 8, 32);
      cnt[v][s] = t;
    }

  const float inv_norm = 1.f / (1.081f * (float)(N_PTS - 1) + 1e-8f);
  if (r == 0) {
#pragma unroll
    for (int v = 0; v < 8; ++v) {
      int i = row0 + v + 8 * half;
#pragma unroll
      for (int s = 0; s < NSH; ++s)
        occ_out[i * NSH + s] = cnt[v][s] * inv_norm;
    }
  }
}

// ---------------------------------------------------------------------------
// Kernel 3: coordination_features = x + occ @ W.T + b   (one thread per elem)
// ---------------------------------------------------------------------------
__global__ __launch_bounds__(256) void feat_kernel(
    const float* __restrict__ x, const float* __restrict__ W,
    const float* __restrict__ b, const float* __restrict__ occ,
    float* __restrict__ out) {
  int idx = blockIdx.x * blockDim.x + threadIdx.x;
  int i = idx >> 8;        // DIM == 256
  int d = idx & (DIM - 1);
  float o0 = occ[i * NSH + 0];
  float o1 = occ[i * NSH + 1];
  float o2 = occ[i * NSH + 2];
  float f = b[d] + o0 * W[d * NSH + 0] + o1 * W[d * NSH + 1] + o2 * W[d * NSH + 2];
  out[idx] = x[idx] + f;
}

// ---------------------------------------------------------------------------
extern "C" void kernel_launch(void* const* d_in, const int* in_sizes, int n_in,
                              void* d_out, int out_size, void* d_ws, size_t ws_size,
                              hipStream_t stream) {
  const float* x   = (const float*)d_in[0];  // (N, D)
  const float* bnd = (const float*)d_in[1];  // (S+1,)
  const float* W   = (const float*)d_in[2];  // (D, S)
  const float* b   = (const float*)d_in[3];  // (D,)

  float* out = (float*)d_out;                     // coordination_features (N*D)
  float* occ = out + (size_t)N_PTS * DIM;         // shell_occupancy (N*S)
  float* sq  = (float*)d_ws;                      // N floats of scratch

  // 1) row norms: 8192 rows, 8 waves/block
  sq_kernel<<<N_PTS / 8, 256, 0, stream>>>(x, sq);
  // 2) WMMA Gram sweep + shell occupancy
  gram_shell_kernel<<<N_PTS / 128, 256, 0, stream>>>(x, sq, bnd, occ);
  // 3) fused feature epilogue
  feat_kernel<<<(N_PTS * DIM) / 256, 256, 0, stream>>>(x, W, b, occ, out);
}